// GAT_78838419685910
// MI455X (gfx1250) — compile-verified
//
#include <hip/hip_runtime.h>
#include <hip/hip_bf16.h>

// ---------------------------------------------------------------------------
// GAT 2-layer forward on gfx1250 (MI455X).
// GEMMs use V_WMMA_F32_16X16X4_F32 (fp32 matrix pipe, wave32, 16x16 tiles).
// Edge softmax / aggregation via global_atomic_add_f32 segment sums.
// ---------------------------------------------------------------------------

typedef __attribute__((ext_vector_type(2))) float v2f;
typedef __attribute__((ext_vector_type(8))) float v8f;

#define GAT_N   50000
#define GAT_E   800000
#define GAT_IN  128
#define GAT_F1  128   // H1*D1
#define GAT_F2  16    // H2*D2
#define GAT_SLOPE 0.2f

// ---------------- WMMA GEMM: C[M,Ncol] = A[M,K] @ B[K,Ncol], fp32 -----------
// One wave per 16x16 tile. blockDim = (32,4): 4 M-tiles per block.
// Requires M % 16 == 0, K % 4 == 0, Ncol % 16 == 0 (true here: 50000/16=3125).
__global__ void gat_wmma_gemm_f32(const float* __restrict__ A,
                                  const float* __restrict__ B,
                                  float* __restrict__ C,
                                  int M, int K, int Ncol) {
    const int mt = blockIdx.x * 4 + threadIdx.y;   // tile row (wave-uniform)
    const int mbase = mt * 16;
    if (mbase >= M) return;                        // uniform across wave
    const int nbase = blockIdx.y * 16;
    const int lane  = threadIdx.x;                 // 0..31 (wave32)
    const int half  = lane >> 4;                   // 0 | 1
    const int l16   = lane & 15;

    const float* __restrict__ arow = A + (size_t)(mbase + l16) * K;
    v8f acc = {};
    for (int k0 = 0; k0 < K; k0 += 4) {
        const int ka = k0 + 2 * half;
        v2f a, b;
        // A 16x4 layout: VGPR0 = K=ka, VGPR1 = K=ka+1 (lanes 0-15: ka=k0, 16-31: ka=k0+2)
        a.x = arow[ka + 0];
        a.y = arow[ka + 1];
        // B 4x16 layout: VGPR v holds row ka+v striped across lanes (cols nbase+l16)
        b.x = B[(size_t)(ka + 0) * Ncol + nbase + l16];
        b.y = B[(size_t)(ka + 1) * Ncol + nbase + l16];
        acc = __builtin_amdgcn_wmma_f32_16x16x4_f32(
            /*neg_a=*/false, a, /*neg_b=*/false, b,
            /*c_mod=*/(short)0, acc, /*reuse_a=*/false, /*reuse_b=*/false);
    }
    // C/D 16x16 f32 layout: VGPR r -> row mbase + r + 8*half, col nbase + l16
    #pragma unroll
    for (int r = 0; r < 8; ++r) {
        C[(size_t)(mbase + r + 8 * half) * Ncol + nbase + l16] = acc[r];
    }
}

// ---------------- zero fill --------------------------------------------------
__global__ void gat_zero(float* __restrict__ p, int n) {
    int t = blockIdx.x * blockDim.x + threadIdx.x;
    if (t < n) p[t] = 0.0f;
}

// ---------------- per-node attention scores (el, er) -------------------------
// One wave per node; lanes reduce over F features; per-head partials reduced
// with wave32 xor-shuffles.
template <int F, int H, int D>
__global__ void gat_scores(const float* __restrict__ ft,
                           const float* __restrict__ al,
                           const float* __restrict__ ar,
                           float* __restrict__ el, float* __restrict__ er,
                           int n) {
    const int node = blockIdx.x * 8 + (threadIdx.x >> 5);
    const int lane = threadIdx.x & 31;
    if (node >= n) return;
    float pl[H], pr[H];
    #pragma unroll
    for (int h = 0; h < H; ++h) { pl[h] = 0.0f; pr[h] = 0.0f; }
    for (int i = lane; i < F; i += 32) {
        const float v = ft[(size_t)node * F + i];
        const int h = i / D;
        pl[h] += v * al[i];
        pr[h] += v * ar[i];
    }
    #pragma unroll
    for (int h = 0; h < H; ++h) {
        float sl = pl[h], sr = pr[h];
        #pragma unroll
        for (int off = 16; off > 0; off >>= 1) {
            sl += __shfl_xor(sl, off, 32);
            sr += __shfl_xor(sr, off, 32);
        }
        if (lane == 0) {
            el[(size_t)node * H + h] = sl;
            er[(size_t)node * H + h] = sr;
        }
    }
}

// ---------------- per-edge: ex = exp(leaky_relu(el[src]+er[dst])), s[dst]+=ex
template <int H>
__global__ void gat_edge_exp(const int* __restrict__ src, const int* __restrict__ dst,
                             const float* __restrict__ el, const float* __restrict__ er,
                             float* __restrict__ ex, float* __restrict__ s, int ne) {
    const int e = blockIdx.x * blockDim.x + threadIdx.x;
    if (e >= ne) return;
    const int si = src[e], di = dst[e];
    #pragma unroll
    for (int h = 0; h < H; ++h) {
        float v = el[(size_t)si * H + h] + er[(size_t)di * H + h];
        v = v > 0.0f ? v : GAT_SLOPE * v;           // leaky_relu
        const float ev = __expf(v);
        ex[(size_t)e * H + h] = ev;
        atomicAdd(&s[(size_t)di * H + h], ev);
    }
}

// ---------------- edge aggregation: out[dst] += (ex/s[dst]) * ft[src] --------
// blockDim = 128; 128/F edges per block; thread -> (edge, feature d).
template <int F, int H, int D>
__global__ void gat_edge_agg(const int* __restrict__ src, const int* __restrict__ dst,
                             const float* __restrict__ ex, const float* __restrict__ s,
                             const float* __restrict__ ft, float* __restrict__ out,
                             int ne) {
    constexpr int EPB = 128 / F;
    const int e = blockIdx.x * EPB + threadIdx.x / F;
    const int d = threadIdx.x % F;
    if (e >= ne) return;
    const int si = src[e], di = dst[e];
    const int h = d / D;
    const float alpha = ex[(size_t)e * H + h] / s[(size_t)di * H + h];
    atomicAdd(&out[(size_t)di * F + d], alpha * ft[(size_t)si * F + d]);
}

// ---------------- alpha output for layer 2 (H2 == 1) -------------------------
__global__ void gat_alpha_out(const int* __restrict__ dst, const float* __restrict__ ex,
                              const float* __restrict__ s, float* __restrict__ aout,
                              int ne) {
    const int e = blockIdx.x * blockDim.x + threadIdx.x;
    if (e >= ne) return;
    aout[e] = ex[e] / s[dst[e]];
}

// ---------------- h1 = elu(agg + b1), in place (F=128) -----------------------
__global__ void gat_elu_bias(float* __restrict__ buf, const float* __restrict__ b,
                             int total) {
    const int t = blockIdx.x * blockDim.x + threadIdx.x;
    if (t >= total) return;
    const float x = buf[t] + b[t & (GAT_F1 - 1)];
    buf[t] = x > 0.0f ? x : (expm1f(x));
}

// ---------------- out += b2 (F=16) -------------------------------------------
__global__ void gat_add_bias16(float* __restrict__ buf, const float* __restrict__ b,
                               int total) {
    const int t = blockIdx.x * blockDim.x + threadIdx.x;
    if (t >= total) return;
    buf[t] += b[t & (GAT_F2 - 1)];
}

static inline int cdiv_i(long long a, long long b) { return (int)((a + b - 1) / b); }

extern "C" void kernel_launch(void* const* d_in, const int* in_sizes, int n_in,
                              void* d_out, int out_size, void* d_ws, size_t ws_size,
                              hipStream_t stream) {
    (void)in_sizes; (void)n_in; (void)out_size; (void)ws_size;
    const int N = GAT_N, E = GAT_E;

    const float* h    = (const float*)d_in[0];
    const int*   src  = (const int*)  d_in[1];
    const int*   dst  = (const int*)  d_in[2];
    const float* W1   = (const float*)d_in[3];
    const float* al1  = (const float*)d_in[4];
    const float* ar1  = (const float*)d_in[5];
    const float* b1   = (const float*)d_in[6];
    const float* W2   = (const float*)d_in[7];
    const float* al2  = (const float*)d_in[8];
    const float* ar2  = (const float*)d_in[9];
    const float* b2   = (const float*)d_in[10];
    const float* Wres = (const float*)d_in[11];

    float* out_logits = (float*)d_out;            // [N, 16]
    float* out_alpha  = out_logits + (size_t)N * GAT_F2;  // [E]

    // workspace carve-up (floats)
    float* w   = (float*)d_ws;
    float* ft1 = w;  w += (size_t)N * GAT_F1;     // [N,128]
    float* h1  = w;  w += (size_t)N * GAT_F1;     // [N,128] agg -> elu output
    float* el1 = w;  w += (size_t)N * 2;
    float* er1 = w;  w += (size_t)N * 2;
    float* s1  = w;  w += (size_t)N * 2;
    float* ex1 = w;  w += (size_t)E * 2;
    float* ft2 = w;  w += (size_t)N * GAT_F2;     // [N,16]
    float* el2 = w;  w += (size_t)N;
    float* er2 = w;  w += (size_t)N;
    float* s2  = w;  w += (size_t)N;
    float* ex2 = w;  w += (size_t)E;

    const dim3 gblk(32, 4);                        // 4 waves / block, 1 tile / wave
    const int  mtiles = N / 16;                    // 3125 (exact)

    // ---- init accumulators ----
    gat_zero<<<cdiv_i((long long)N * 2, 256), 256, 0, stream>>>(s1, N * 2);
    gat_zero<<<cdiv_i((long long)N * GAT_F1, 256), 256, 0, stream>>>(h1, N * GAT_F1);
    gat_zero<<<cdiv_i(N, 256), 256, 0, stream>>>(s2, N);

    // ---- layer 1 ----
    gat_wmma_gemm_f32<<<dim3(cdiv_i(mtiles, 4), GAT_F1 / 16), gblk, 0, stream>>>(
        h, W1, ft1, N, GAT_IN, GAT_F1);
    gat_scores<GAT_F1, 2, 64><<<cdiv_i(N, 8), 256, 0, stream>>>(ft1, al1, ar1, el1, er1, N);
    gat_edge_exp<2><<<cdiv_i(E, 256), 256, 0, stream>>>(src, dst, el1, er1, ex1, s1, E);
    gat_edge_agg<GAT_F1, 2, 64><<<E, 128, 0, stream>>>(src, dst, ex1, s1, ft1, h1, E);
    gat_elu_bias<<<cdiv_i((long long)N * GAT_F1, 256), 256, 0, stream>>>(h1, b1, N * GAT_F1);

    // ---- layer 2 ----
    gat_wmma_gemm_f32<<<dim3(cdiv_i(mtiles, 4), GAT_F2 / 16), gblk, 0, stream>>>(
        h1, W2, ft2, N, GAT_F1, GAT_F2);
    gat_wmma_gemm_f32<<<dim3(cdiv_i(mtiles, 4), GAT_F2 / 16), gblk, 0, stream>>>(
        h1, Wres, out_logits, N, GAT_F1, GAT_F2);          // res into d_out
    gat_add_bias16<<<cdiv_i((long long)N * GAT_F2, 256), 256, 0, stream>>>(
        out_logits, b2, N * GAT_F2);
    gat_scores<GAT_F2, 1, 16><<<cdiv_i(N, 8), 256, 0, stream>>>(ft2, al2, ar2, el2, er2, N);
    gat_edge_exp<1><<<cdiv_i(E, 256), 256, 0, stream>>>(src, dst, el2, er2, ex2, s2, E);
    gat_alpha_out<<<cdiv_i(E, 256), 256, 0, stream>>>(dst, ex2, s2, out_alpha, E);
    gat_edge_agg<GAT_F2, 1, 16><<<cdiv_i(E, 8), 128, 0, stream>>>(
        src, dst, ex2, s2, ft2, out_logits, E);
    // logits.mean(axis=1) with H2==1 is identity -> d_out already final.
}